// RandomMultiContrastRemap_37288906063999
// MI455X (gfx1250) — compile-verified
//
#include <hip/hip_runtime.h>
#include <hip/hip_bf16.h>
#include <stdint.h>

// ---------------- problem constants ----------------
#define DD        160
#define NTOT      (160 * 160 * 160)
#define NBINS     128
#define GSIGMA    12          // NUM_BINS // 10
#define GK        73          // 6*SIGMA+1 | 1
#define EM_ITERS  40
#define FLOOD_ITERS 160       // >= max Chebyshev geodesic from border (outside of convex body)

typedef unsigned char u8;
typedef __attribute__((ext_vector_type(2))) float v2f;
typedef __attribute__((ext_vector_type(8))) float v8f;

struct SelState {
  unsigned prefix[2];
  unsigned kth[2];
  float    frac;
};

// ---------------- device helpers ----------------
__device__ __forceinline__ unsigned f2k(float f) {
  unsigned u = __float_as_uint(f);
  return (u & 0x80000000u) ? ~u : (u | 0x80000000u);   // order-preserving float->uint
}
__device__ __forceinline__ float k2f(unsigned k) {
  unsigned u = (k & 0x80000000u) ? (k ^ 0x80000000u) : ~k;
  return __uint_as_float(u);
}

// Threefry-2x32, 20 rounds (matches JAX's threefry2x32)
__device__ __forceinline__ void threefry2x32(unsigned k0, unsigned k1,
                                             unsigned x0, unsigned x1,
                                             unsigned* o0, unsigned* o1) {
  const unsigned ks2 = 0x1BD11BDAu ^ k0 ^ k1;
  const int r0[4] = {13, 15, 26, 6};
  const int r1[4] = {17, 29, 16, 24};
  x0 += k0; x1 += k1;
#pragma unroll
  for (int g = 0; g < 5; ++g) {
    const int* R = (g & 1) ? r1 : r0;
#pragma unroll
    for (int r = 0; r < 4; ++r) {
      x0 += x1;
      x1 = (x1 << R[r]) | (x1 >> (32 - R[r]));
      x1 ^= x0;
    }
    unsigned a, b;
    switch (g) {
      case 0: a = k1;  b = ks2; break;
      case 1: a = ks2; b = k0;  break;
      case 2: a = k0;  b = k1;  break;
      case 3: a = k1;  b = ks2; break;
      default: a = ks2; b = k0; break;
    }
    x0 += a; x1 += b + (unsigned)(g + 1);
  }
  *o0 = x0; *o1 = x1;
}

// ---------------- morphology ----------------
__global__ void rcr_u8_from_bool(const u8* __restrict__ src, u8* __restrict__ dst, int n) {
  int i = blockIdx.x * blockDim.x + threadIdx.x;
  if (i >= n) return;
  dst[i] = src[i] ? (u8)1 : (u8)0;
}

// separable clipped-window OR (isAnd=0) / AND (isAnd=1) along one axis
__global__ void rcr_pool1d(const u8* __restrict__ src, u8* __restrict__ dst,
                           int stride, int len, int radius, int isAnd) {
  int i = blockIdx.x * blockDim.x + threadIdx.x;
  if (i >= NTOT) return;
  int c  = (i / stride) % len;
  int lo = c - radius; if (lo < 0) lo = 0;
  int hi = c + radius; if (hi > len - 1) hi = len - 1;
  u8 v = isAnd ? (u8)1 : (u8)0;
  for (int cc = lo; cc <= hi; ++cc) {
    u8 s = src[i + (cc - c) * stride];
    v = isAnd ? (u8)(v & s) : (u8)(v | s);
  }
  dst[i] = v;
}

__global__ void rcr_inv_border(const u8* __restrict__ closed, u8* __restrict__ inv,
                               u8* __restrict__ curr, int n) {
  int i = blockIdx.x * blockDim.x + threadIdx.x;
  if (i >= n) return;
  int x = i % DD, y = (i / DD) % DD, z = i / (DD * DD);
  u8 ic = closed[i] ? (u8)0 : (u8)1;
  inv[i] = ic;
  bool face = (x == 0) || (x == DD - 1) || (y == 0) || (y == DD - 1) || (z == 0) || (z == DD - 1);
  curr[i] = (ic && face) ? (u8)1 : (u8)0;
}

__global__ void rcr_flood_xy(const u8* __restrict__ curr, u8* __restrict__ tmp, int n) {
  int i = blockIdx.x * blockDim.x + threadIdx.x;
  if (i >= n) return;
  int x = i % DD, y = (i / DD) % DD;
  u8 v = 0;
#pragma unroll
  for (int dy = -1; dy <= 1; ++dy) {
    int yy = y + dy; if (yy < 0 || yy >= DD) continue;
#pragma unroll
    for (int dx = -1; dx <= 1; ++dx) {
      int xx = x + dx; if (xx < 0 || xx >= DD) continue;
      v |= curr[i + dy * DD + dx];
    }
  }
  tmp[i] = v;
}

__global__ void rcr_flood_z(const u8* __restrict__ tmp, const u8* __restrict__ inv,
                            u8* __restrict__ next, int n) {
  int i = blockIdx.x * blockDim.x + threadIdx.x;
  if (i >= n) return;
  int z = i / (DD * DD);
  u8 v = 0;
#pragma unroll
  for (int dz = -1; dz <= 1; ++dz) {
    int zz = z + dz; if (zz < 0 || zz >= DD) continue;
    v |= tmp[i + dz * DD * DD];
  }
  next[i] = (u8)(v & inv[i]);
}

// m3 = closed | (inv & ~reached) == !(inv && reached)
__global__ void rcr_make_m3(const u8* __restrict__ inv, const u8* __restrict__ reached,
                            u8* __restrict__ m3, int n) {
  int i = blockIdx.x * blockDim.x + threadIdx.x;
  if (i >= n) return;
  m3[i] = (inv[i] && reached[i]) ? (u8)0 : (u8)1;
}

// ---------------- small state / reductions ----------------
__global__ void rcr_scal_init(unsigned* scal, float* gaccum) {
  if (blockIdx.x || threadIdx.x) return;
  scal[0] = 0u;            // masked count
  scal[1] = (unsigned)NTOT;
  for (int r = 0; r < 16; ++r) gaccum[r] = 0.0f;
}

__global__ __launch_bounds__(256) void rcr_count(const u8* __restrict__ m3, unsigned* cnt, int n) {
  __shared__ unsigned sh[256];
  unsigned c = 0;
  for (int i = blockIdx.x * blockDim.x + threadIdx.x; i < n; i += gridDim.x * blockDim.x)
    c += (unsigned)m3[i];
  sh[threadIdx.x] = c;
  __syncthreads();
  for (int s = 128; s > 0; s >>= 1) {
    if (threadIdx.x < s) sh[threadIdx.x] += sh[threadIdx.x + s];
    __syncthreads();
  }
  if (threadIdx.x == 0) atomicAdd(cnt, sh[0]);
}

// ---------------- radix select (exact k-th order statistic, device-driven) ----------------
__global__ void rcr_sel_init(SelState* s, const unsigned* nptr, float q, unsigned* hist) {
  if (blockIdx.x || threadIdx.x) return;
  for (int j = 0; j < 512; ++j) hist[j] = 0u;
  unsigned n = *nptr;
  double pos = (double)q * (double)(n - 1u);
  unsigned k0 = (unsigned)pos;
  if (k0 > n - 1u) k0 = n - 1u;
  s->frac = (float)(pos - (double)k0);
  s->kth[0] = k0;
  s->kth[1] = (k0 + 1u <= n - 1u) ? (k0 + 1u) : (n - 1u);
  s->prefix[0] = 0u;
  s->prefix[1] = 0u;
}

__global__ __launch_bounds__(256) void rcr_sel_hist(const float* __restrict__ data,
                                                    const u8* __restrict__ msk,
                                                    const SelState* __restrict__ s,
                                                    unsigned* __restrict__ hist,
                                                    int byte, int n) {
  __shared__ unsigned lh[512];
  for (int j = threadIdx.x; j < 512; j += 256) lh[j] = 0u;
  __syncthreads();
  const unsigned p0 = s->prefix[0], p1 = s->prefix[1];
  const unsigned above = (byte == 3) ? 0u : (0xFFFFFFFFu << (8 * (byte + 1)));
  const int shift = 8 * byte;
  const int stride = gridDim.x * blockDim.x;
  for (int i = blockIdx.x * blockDim.x + threadIdx.x; i < n; i += stride) {
    __builtin_prefetch(&data[i + 4 * stride], 0, 1);   // global_prefetch_b8
    if (msk && !msk[i]) continue;
    unsigned key = f2k(data[i]);
    unsigned bin = (key >> shift) & 255u;
    if ((key & above) == (p0 & above)) atomicAdd(&lh[bin], 1u);
    if ((key & above) == (p1 & above)) atomicAdd(&lh[256 + bin], 1u);
  }
  __syncthreads();
  for (int j = threadIdx.x; j < 512; j += 256)
    if (lh[j]) atomicAdd(&hist[j], lh[j]);
}

__global__ void rcr_sel_step(SelState* s, unsigned* hist, int byte, float* qdst) {
  if (blockIdx.x || threadIdx.x) return;
  float v[2] = {0.f, 0.f};
  for (int st = 0; st < 2; ++st) {
    unsigned kth = s->kth[st];
    unsigned cum = 0;
    int chosen = -1;
    for (int b = 0; b < 256; ++b) {
      unsigned c = hist[st * 256 + b];
      if (chosen < 0 && cum + c > kth) { chosen = b; s->kth[st] = kth - cum; }
      cum += c;
      hist[st * 256 + b] = 0u;   // ready for next pass
    }
    if (chosen < 0) chosen = 255;
    s->prefix[st] |= ((unsigned)chosen) << (8 * byte);
    if (byte == 0) v[st] = k2f(s->prefix[st]);
  }
  if (byte == 0 && qdst) *qdst = v[0] + s->frac * (v[1] - v[0]);  // linear interpolation
}

__global__ void rcr_centers(const float* qv, const unsigned* scal, float* centers, float* wsumf) {
  if (blockIdx.x || threadIdx.x) return;
  float a = qv[0], b = qv[1];
  centers[0] = a;
  centers[1] = 0.5f * (a + b);
  centers[2] = b;
  *wsumf = (float)scal[0];
}

// ---------------- GMM accumulate: E-step on VALU, M-step sums reduced on the WMMA pipe --------
// acc rows: [0..2]=Sum p_c, [3..5]=Sum p_c*x, [6..8]=Sum p_c*x^2   (w==1 on masked voxels)
__global__ __launch_bounds__(256) void rcr_accum(const float* __restrict__ x,
                                                 const u8* __restrict__ m3,
                                                 const float* __restrict__ pvec,
                                                 float* __restrict__ gaccum,
                                                 int mode, int n) {
  __shared__ float red[16][260];
  float acc[9];
#pragma unroll
  for (int j = 0; j < 9; ++j) acc[j] = 0.0f;

  float c0, c1, c2, m0, m1, m2, iv0, iv1, iv2;
  if (mode == 0) {               // hard assignment to centers
    c0 = pvec[0]; c1 = pvec[1]; c2 = pvec[2];
    m0 = m1 = m2 = iv0 = iv1 = iv2 = 0.f;
  } else {                       // soft E-step: coef_c = pi_c / sqrt(2*pi*var_c)
    const float TWO_PI = 6.283185307179586f;
    float v0 = pvec[6], v1 = pvec[7], v2 = pvec[8];
    c0 = pvec[0] * rsqrtf(TWO_PI * v0);
    c1 = pvec[1] * rsqrtf(TWO_PI * v1);
    c2 = pvec[2] * rsqrtf(TWO_PI * v2);
    m0 = pvec[3]; m1 = pvec[4]; m2 = pvec[5];
    iv0 = -0.5f / v0; iv1 = -0.5f / v1; iv2 = -0.5f / v2;
  }

  const int stride = gridDim.x * blockDim.x;
  for (int i = blockIdx.x * blockDim.x + threadIdx.x; i < n; i += stride) {
    __builtin_prefetch(&x[i + 4 * stride], 0, 1);
    if (!m3[i]) continue;
    float xv = x[i];
    float p0, p1, p2;
    if (mode == 0) {
      float d0 = fabsf(xv - c0), d1 = fabsf(xv - c1), d2 = fabsf(xv - c2);
      int lbl = 0; float best = d0;
      if (d1 < best) { best = d1; lbl = 1; }
      if (d2 < best) { lbl = 2; }
      p0 = (lbl == 0) ? 1.f : 0.f;
      p1 = (lbl == 1) ? 1.f : 0.f;
      p2 = (lbl == 2) ? 1.f : 0.f;
    } else {
      float d0 = xv - m0, d1 = xv - m1, d2 = xv - m2;
      p0 = fmaxf(c0 * __expf(iv0 * d0 * d0), 1e-12f);
      p1 = fmaxf(c1 * __expf(iv1 * d1 * d1), 1e-12f);
      p2 = fmaxf(c2 * __expf(iv2 * d2 * d2), 1e-12f);
      float inv = 1.0f / (p0 + p1 + p2);
      p0 *= inv; p1 *= inv; p2 *= inv;
    }
    acc[0] += p0; acc[1] += p1; acc[2] += p2;
    acc[3] += p0 * xv; acc[4] += p1 * xv; acc[5] += p2 * xv;
    float x2 = xv * xv;
    acc[6] += p0 * x2; acc[7] += p1 * x2; acc[8] += p2 * x2;
  }

  const int tid = threadIdx.x;
#pragma unroll
  for (int j = 0; j < 16; ++j) red[j][tid] = (j < 9) ? acc[j] : 0.0f;
  __syncthreads();

  // Row-sum of the 16x256 partials matrix on the matrix pipe:
  // A tile = 16x4 chunk (documented 16x4 f32 A layout), B = ones(4x16), chained f32 accumulate.
  if (tid < 32) {               // whole wave 0 -> EXEC all ones for WMMA
    v8f d = {0.f, 0.f, 0.f, 0.f, 0.f, 0.f, 0.f, 0.f};
    v2f bones; bones.x = 1.0f; bones.y = 1.0f;
    const int m  = tid & 15;
    const int kb = (tid >> 4) << 1;      // lanes 0-15 -> K{0,1}; lanes 16-31 -> K{2,3}
    for (int c = 0; c < 64; ++c) {
      v2f a;
      a.x = red[m][4 * c + kb];
      a.y = red[m][4 * c + kb + 1];
      d = __builtin_amdgcn_wmma_f32_16x16x4_f32(false, a, false, bones, (short)0, d, false, false);
    }
    // D layout: vgpr r, lanes 0-15 -> row r; lanes 16-31 -> row r+8
    if (tid == 0) {
#pragma unroll
      for (int r = 0; r < 8; ++r) atomicAdd(&gaccum[r], d[r]);
    } else if (tid == 16) {
#pragma unroll
      for (int r = 0; r < 8; ++r) atomicAdd(&gaccum[8 + r], d[r]);
    }
  }
}

__global__ void rcr_finalize(float* gaccum, float* params, const float* wsumf, int mode) {
  if (blockIdx.x || threadIdx.x) return;
  float ws = *wsumf;
  for (int c = 0; c < 3; ++c) {
    float S0 = gaccum[c], S1 = gaccum[3 + c], S2 = gaccum[6 + c];
    float Nk = S0;
    float pi = Nk / ws;
    float mu = S1 / Nk;
    float num = S2 - 2.0f * mu * S1 + mu * mu * S0;   // Sum p*(x-mu)^2
    float var;
    if (mode == 0) {
      float den = Nk - 1.0f; if (den < 1.0f) den = 1.0f;
      var = fmaxf(num / den, 1e-4f);
    } else {
      var = fmaxf(num / Nk, 1e-4f);
    }
    params[c] = pi; params[3 + c] = mu; params[6 + c] = var;
  }
  for (int r = 0; r < 16; ++r) gaccum[r] = 0.0f;      // ready for next accumulate
}

__global__ void rcr_sort_params(float* p) {   // post[:, argsort(mu)]
  if (blockIdx.x || threadIdx.x) return;
  float pi[3] = {p[0], p[1], p[2]}, mu[3] = {p[3], p[4], p[5]}, va[3] = {p[6], p[7], p[8]};
  int idx[3] = {0, 1, 2};
  for (int a = 0; a < 2; ++a)
    for (int b = 0; b < 2 - a; ++b)
      if (mu[idx[b]] > mu[idx[b + 1]]) { int t = idx[b]; idx[b] = idx[b + 1]; idx[b + 1] = t; }
  for (int c = 0; c < 3; ++c) {
    p[c] = pi[idx[c]]; p[3 + c] = mu[idx[c]]; p[6 + c] = va[idx[c]];
  }
}

// mapped = x * sum_c sqrt(post_c) * TARGETS[c]; x2 = masked ? mapped : x
__global__ void rcr_map(const float* __restrict__ x, const u8* __restrict__ m3,
                        const float* __restrict__ params, float* __restrict__ out0, int n) {
  int i = blockIdx.x * blockDim.x + threadIdx.x;
  if (i >= n) return;
  const float TWO_PI = 6.283185307179586f;
  float v0 = params[6], v1 = params[7], v2 = params[8];
  float c0 = params[0] * rsqrtf(TWO_PI * v0);
  float c1 = params[1] * rsqrtf(TWO_PI * v1);
  float c2 = params[2] * rsqrtf(TWO_PI * v2);
  float m0 = params[3], m1 = params[4], m2 = params[5];
  float xv = x[i];
  float d0 = xv - m0, d1 = xv - m1, d2 = xv - m2;
  float p0 = fmaxf(c0 * __expf(-0.5f * d0 * d0 / v0), 1e-12f);
  float p1 = fmaxf(c1 * __expf(-0.5f * d1 * d1 / v1), 1e-12f);
  float p2 = fmaxf(c2 * __expf(-0.5f * d2 * d2 / v2), 1e-12f);
  float inv = 1.0f / (p0 + p1 + p2);
  float mapped = xv * (sqrtf(p0 * inv) * 4.5f + sqrtf(p1 * inv) * 0.325f + sqrtf(p2 * inv) * 0.2f);
  out0[i] = m3[i] ? mapped : xv;
}

// ---------------- random LUT (Threefry key=(0,1), exactly JAX's uniform) ----------------
__global__ void rcr_lut(float* __restrict__ lutOut) {
  __shared__ float lut[NBINS];
  __shared__ float g[GK];
  __shared__ float sm[NBINS];
  __shared__ float stats[4];
  int t = threadIdx.x;
  if (t < 64) {
    unsigned o0, o1;
    threefry2x32(0u, 1u, (unsigned)t, (unsigned)(t + 64), &o0, &o1);
    lut[t]      = __uint_as_float((o0 >> 9) | 0x3F800000u) - 1.0f;
    lut[t + 64] = __uint_as_float((o1 >> 9) | 0x3F800000u) - 1.0f;
  }
  if (t < GK) {
    float xx = (float)(t - GK / 2) / (float)GSIGMA;
    g[t] = __expf(-0.5f * xx * xx);
  }
  __syncthreads();
  if (t == 0) {
    float gs = 0.f;
    for (int i = 0; i < GK; ++i) gs += g[i];
    stats[0] = gs;
  }
  __syncthreads();
  float ginv = 1.0f / stats[0];
  {   // 'same' convolution: s[i] = sum_j lut[j] * g[i + 36 - j]
    float acc = 0.f;
    int jlo = t - GK / 2; if (jlo < 0) jlo = 0;
    int jhi = t + GK / 2; if (jhi > NBINS - 1) jhi = NBINS - 1;
    for (int j = jlo; j <= jhi; ++j) acc += lut[j] * g[t + GK / 2 - j];
    sm[t] = acc * ginv;
  }
  __syncthreads();
  if (t == 0) {
    float mn = sm[0], mx = sm[0];
    for (int i = 1; i < NBINS; ++i) { mn = fminf(mn, sm[i]); mx = fmaxf(mx, sm[i]); }
    stats[1] = mn; stats[2] = mx;
  }
  __syncthreads();
  float v = (sm[t] - stats[1]) / (stats[2] - stats[1]);
  if (t == 0) v = 0.0f;
  if (t == NBINS - 1) v = 1.0f;
  lutOut[t] = v;
}

// ---------------- final: normalize, LUT interp, mask ----------------
__global__ void rcr_final(const float* __restrict__ out0, const u8* __restrict__ m3,
                          const float* __restrict__ lut, const float* __restrict__ qv,
                          float* __restrict__ out, int n) {
  int i = blockIdx.x * blockDim.x + threadIdx.x;
  if (i >= n) return;
  float vmin = qv[2], vmax = qv[3];
  float v = (out0[i] - vmin) / (vmax - vmin);
  v = fminf(fmaxf(v, 0.0f), 1.0f);
  float idx = v * (float)(NBINS - 1);
  int lo = (int)floorf(idx);
  if (lo < 0) lo = 0;
  if (lo > NBINS - 2) lo = NBINS - 2;
  float wq = idx - (float)lo;
  float res = lut[lo] + (lut[lo + 1] - lut[lo]) * wq;
  out[i] = m3[i] ? res : 0.0f;
}

// ---------------- host orchestration ----------------
extern "C" void kernel_launch(void* const* d_in, const int* in_sizes, int n_in,
                              void* d_out, int out_size, void* d_ws, size_t ws_size,
                              hipStream_t stream) {
  (void)in_sizes; (void)n_in; (void)out_size; (void)ws_size;
  const float* img    = (const float*)d_in[0];
  const u8*    maskIn = (const u8*)d_in[1];    // numpy bool = 1 byte
  float*       out    = (float*)d_out;

  char* ws = (char*)d_ws;
  size_t off = 0;
  auto alloc = [&](size_t b) { size_t o = off; off += (b + 255) & ~(size_t)255; return o; };
  u8*       A      = (u8*)(ws + alloc(NTOT));
  u8*       Bb     = (u8*)(ws + alloc(NTOT));
  u8*       Cc     = (u8*)(ws + alloc(NTOT));
  float*    out0   = (float*)(ws + alloc((size_t)NTOT * sizeof(float)));
  float*    gaccum = (float*)(ws + alloc(16 * sizeof(float)));
  float*    params = (float*)(ws + alloc(16 * sizeof(float)));
  float*    centers= (float*)(ws + alloc(4 * sizeof(float)));
  float*    qv     = (float*)(ws + alloc(4 * sizeof(float)));
  float*    wsumf  = (float*)(ws + alloc(4 * sizeof(float)));
  unsigned* scal   = (unsigned*)(ws + alloc(4 * sizeof(unsigned)));
  unsigned* hist   = (unsigned*)(ws + alloc(512 * sizeof(unsigned)));
  SelState* sel    = (SelState*)(ws + alloc(sizeof(SelState)));
  float*    lutBuf = (float*)(ws + alloc(NBINS * sizeof(float)));

  const int B  = 256;
  const int GE = (NTOT + B - 1) / B;   // elementwise grid
  const int GR = 2048;                 // grid-stride reduction grid

  // ---- fill_holes(mask, 5) ----
  rcr_u8_from_bool<<<GE, B, 0, stream>>>(maskIn, A, NTOT);
  rcr_pool1d<<<GE, B, 0, stream>>>(A, Bb, 1,       DD, 2, 0);   // dilate x
  rcr_pool1d<<<GE, B, 0, stream>>>(Bb, Cc, DD,     DD, 2, 0);   // dilate y
  rcr_pool1d<<<GE, B, 0, stream>>>(Cc, A, DD * DD, DD, 2, 0);   // dilate z   -> dil in A
  rcr_pool1d<<<GE, B, 0, stream>>>(A, Bb, 1,       DD, 2, 1);   // erode x
  rcr_pool1d<<<GE, B, 0, stream>>>(Bb, Cc, DD,     DD, 2, 1);   // erode y
  rcr_pool1d<<<GE, B, 0, stream>>>(Cc, Bb, DD * DD, DD, 2, 1);  // erode z    -> closed in Bb
  rcr_inv_border<<<GE, B, 0, stream>>>(Bb, Cc, A, NTOT);        // inv=Cc, seeds=A
  for (int it = 0; it < FLOOD_ITERS; ++it) {                    // fixed-point flood fill
    rcr_flood_xy<<<GE, B, 0, stream>>>(A, Bb, NTOT);
    rcr_flood_z<<<GE, B, 0, stream>>>(Bb, Cc, A, NTOT);
  }
  rcr_make_m3<<<GE, B, 0, stream>>>(Cc, A, Bb, NTOT);           // m3 = Bb
  u8* m3 = Bb;

  // ---- counts / wsum ----
  rcr_scal_init<<<1, 1, 0, stream>>>(scal, gaccum);
  rcr_count<<<GR, B, 0, stream>>>(m3, &scal[0], NTOT);

  // ---- nanquantile(x[mask], .05/.95) via exact radix select ----
  rcr_sel_init<<<1, 1, 0, stream>>>(sel, &scal[0], 0.05f, hist);
  for (int by = 3; by >= 0; --by) {
    rcr_sel_hist<<<GR, B, 0, stream>>>(img, m3, sel, hist, by, NTOT);
    rcr_sel_step<<<1, 1, 0, stream>>>(sel, hist, by, (by == 0) ? &qv[0] : (float*)nullptr);
  }
  rcr_sel_init<<<1, 1, 0, stream>>>(sel, &scal[0], 0.95f, hist);
  for (int by = 3; by >= 0; --by) {
    rcr_sel_hist<<<GR, B, 0, stream>>>(img, m3, sel, hist, by, NTOT);
    rcr_sel_step<<<1, 1, 0, stream>>>(sel, hist, by, (by == 0) ? &qv[1] : (float*)nullptr);
  }
  rcr_centers<<<1, 1, 0, stream>>>(qv, scal, centers, wsumf);

  // ---- GMM init (hard labels) + 39 EM steps ----
  rcr_accum<<<GR, B, 0, stream>>>(img, m3, centers, gaccum, 0, NTOT);
  rcr_finalize<<<1, 1, 0, stream>>>(gaccum, params, wsumf, 0);
  for (int it = 0; it < EM_ITERS - 1; ++it) {
    rcr_accum<<<GR, B, 0, stream>>>(img, m3, params, gaccum, 1, NTOT);
    rcr_finalize<<<1, 1, 0, stream>>>(gaccum, params, wsumf, 1);
  }
  rcr_sort_params<<<1, 1, 0, stream>>>(params);

  // ---- fuzzy remap ----
  rcr_map<<<GE, B, 0, stream>>>(img, m3, params, out0, NTOT);

  // ---- quantile(out, .05/.95) over all voxels ----
  rcr_sel_init<<<1, 1, 0, stream>>>(sel, &scal[1], 0.05f, hist);
  for (int by = 3; by >= 0; --by) {
    rcr_sel_hist<<<GR, B, 0, stream>>>(out0, (const u8*)nullptr, sel, hist, by, NTOT);
    rcr_sel_step<<<1, 1, 0, stream>>>(sel, hist, by, (by == 0) ? &qv[2] : (float*)nullptr);
  }
  rcr_sel_init<<<1, 1, 0, stream>>>(sel, &scal[1], 0.95f, hist);
  for (int by = 3; by >= 0; --by) {
    rcr_sel_hist<<<GR, B, 0, stream>>>(out0, (const u8*)nullptr, sel, hist, by, NTOT);
    rcr_sel_step<<<1, 1, 0, stream>>>(sel, hist, by, (by == 0) ? &qv[3] : (float*)nullptr);
  }

  // ---- random contrast LUT + final ----
  rcr_lut<<<1, NBINS, 0, stream>>>(lutBuf);
  rcr_final<<<GE, B, 0, stream>>>(out0, m3, lutBuf, qv, out, NTOT);
}